// ReUNetModel_29592324669534
// MI455X (gfx1250) — compile-verified
//
#include <hip/hip_runtime.h>

// Gaussian-weighted local AdaIN (B=4, C=64, H=W=512, sigma=1, k=7, fp32).
// One wave computes a 32x16 output tile. Both separable blur passes are
// banded-matrix matmuls on V_WMMA_F32_16X16X4_F32 (fp32 accumulate).
// Both input windows are staged to LDS up front for max memory-level
// parallelism; all address math lands on SALU + instruction immediates.

typedef __attribute__((ext_vector_type(2))) float v2f;
typedef __attribute__((ext_vector_type(8))) float v8f;

// Normalized Gaussian taps for sigma=1, k=7: band value g[d], 0 outside [0,6].
__device__ __forceinline__ float gband(int d) {
    float r = 0.0f;
    r = (d == 0 || d == 6) ? 0.00443304818f : r;
    r = (d == 1 || d == 5) ? 0.05400558263f : r;
    r = (d == 2 || d == 4) ? 0.24203622937f : r;
    r = (d == 3)           ? 0.39905030374f : r;
    return r;
}

// jnp.pad mode='reflect' index mapping for extent 512 (pad <= 3).
__device__ __forceinline__ int reflect512(int i) {
    i = (i < 0) ? -i : i;
    return (i >= 512) ? (1022 - i) : i;
}

__device__ __forceinline__ v8f wmma4(v2f a, v2f b, v8f c) {
    // D(16x16,f32) = A(16x4,f32) * B(4x16,f32) + C
    return __builtin_amdgcn_wmma_f32_16x16x4_f32(false, a, false, b, (short)0, c,
                                                 false, false);
}

// Stage a 38x22 reflect-padded window (lane = column) into LDS.
__device__ __forceinline__ void stage_window(const float* __restrict__ img,
                                             float* __restrict__ xb,
                                             int row0, int gc, int lane) {
    if (lane < 22) {
        if (row0 >= 32 && row0 <= 448) {
            // Interior: one base pointer, row offsets become load immediates.
            const float* rp = img + ((row0 - 3) << 9) + gc;
#pragma unroll
            for (int r = 0; r < 38; ++r)
                xb[r * 25 + lane] = rp[r << 9];
        } else {
            // Border: reflect the (wave-uniform) row index on the scalar unit.
#pragma unroll
            for (int r = 0; r < 38; ++r) {
                int gr = reflect512(row0 - 3 + r);
                xb[r * 25 + lane] = img[(gr << 9) + gc];
            }
        }
    }
}

// Banded separable blur: mean / second moment for row-chunks oc=0,1 (16x16
// each) in WMMA C/D layout. xb: 38x25 window, tb: 40x32 T scratch.
__device__ __forceinline__ void compute_stats(
    const float* __restrict__ xb, float* __restrict__ tb,
    const v2f (&wc)[6], int n, int h, int off5,
    v8f& mean0, v8f& m20, v8f& mean1, v8f& m21) {
    // ---- Pass 1 (horizontal): T[r][n] = sum_j X[r][j] g[j-n], rows 0..37 ----
#pragma unroll
    for (int mr = 0; mr < 3; ++mr) {
        int rr = 16 * mr + n;
        rr = rr > 37 ? 37 : rr;            // rows 38..47 hit zero V-band later
        const float* rowp = xb + rr * 25;
        const float* ap = rowp + 2 * h;
        v8f acc1 = {};
        v8f acc2 = {};
#pragma unroll
        for (int kc = 0; kc < 5; ++kc) {   // ds_load_2addr, immediate offsets
            v2f a;
            a[0] = ap[4 * kc];
            a[1] = ap[4 * kc + 1];
            v2f a2 = a * a;
            acc1 = wmma4(a, wc[kc], acc1);
            acc2 = wmma4(a2, wc[kc], acc2);
        }
        {   // kc = 5: hi half would read cols 22,23 -> clamp to col 21 (x 0-band)
            v2f a;
            a[0] = rowp[off5];             // off5 = h ? 21 : 20
            a[1] = rowp[21];
            v2f a2 = a * a;
            acc1 = wmma4(a, wc[5], acc1);
            acc2 = wmma4(a2, wc[5], acc2);
        }
        // Store D rows 16*mr + v + 8*h into T (keep rows < 40).
        if (mr < 2 || h == 0) {
            float* tp = tb + (16 * mr + 8 * h) * 32 + n;
#pragma unroll
            for (int v = 0; v < 8; ++v) {
                tp[v * 32] = acc1[v];      // moment 1 at col n
                tp[v * 32 + 16] = acc2[v]; // moment 2 at col 16+n
            }
        }
    }
    __builtin_amdgcn_wave_barrier();

    // ---- Pass 2 (vertical): Y[m][n] = sum_j g[j-m] T[j][n], per row-chunk ----
#pragma unroll
    for (int oc = 0; oc < 2; ++oc) {
        v8f mm = {};
        v8f qq = {};
        const float* bp = tb + (16 * oc + 2 * h) * 32 + n;
#pragma unroll
        for (int kc = 0; kc < 6; ++kc) {
            v2f b1, b2;
            b1[0] = bp[kc * 128];
            b1[1] = bp[kc * 128 + 32];
            b2[0] = bp[kc * 128 + 16];
            b2[1] = bp[kc * 128 + 48];
            mm = wmma4(wc[kc], b1, mm);
            qq = wmma4(wc[kc], b2, qq);
        }
        if (oc == 0) { mean0 = mm; m20 = qq; }
        else         { mean1 = mm; m21 = qq; }
    }
}

__global__ __launch_bounds__(256) void adain_local_kernel(
    const float* __restrict__ content, const float* __restrict__ style,
    float* __restrict__ out, int total_tiles) {
    __shared__ float smem[3200 * 8];       // per wave: 950 + 950 (xb) + 1280 (tb)
    const int lane = threadIdx.x & 31;
    const int wave = threadIdx.x >> 5;
    float* xb_c = smem + wave * 3200;
    float* xb_s = xb_c + 960;
    float* tb = xb_s + 960;

    // Wave-uniform tile id in an SGPR so all tile math lands on the SALU.
    const int tile = __builtin_amdgcn_readfirstlane((int)blockIdx.x * 8 + wave);
    if (tile >= total_tiles) return;

    const int plane = tile >> 9;           // 16x32 tiles of 32x16 per plane
    const int tin = tile & 511;
    const int row0 = (tin >> 5) << 5;      // 0,32,...,480
    const int col0 = (tin & 31) << 4;      // 0,16,...,496
    const long base = (long)plane << 18;   // plane * 512 * 512

    const int n = lane & 15;
    const int h = lane >> 4;
    const int off5 = h ? 21 : 20;

    // Banded Gaussian coefficients: lane holds g[(4kc+2h+{0,1}) - (lane&15)].
    // Serves as B (pass-1 W matrix) and A (pass-2 V matrix) for both images.
    // Pin the values in VGPRs so the compiler cannot rematerialize the select
    // chains inside the WMMA loops.
    v2f wc[6];
#pragma unroll
    for (int kc = 0; kc < 6; ++kc) {
        int j = 4 * kc + 2 * h;
        float w0 = gband(j - n);
        float w1 = gband(j + 1 - n);
        asm("" : "+v"(w0), "+v"(w1));
        wc[kc][0] = w0;
        wc[kc][1] = w1;
    }

    // Reflected global column for this staging lane (computed once).
    const int gc = reflect512(col0 - 3 + (lane < 22 ? lane : 21));

    // Stage both windows up front: 76 loads in flight before any math.
    stage_window(content + base, xb_c, row0, gc, lane);
    stage_window(style + base, xb_s, row0, gc, lane);
    __builtin_amdgcn_wave_barrier();

    v8f cme0, cm20, cme1, cm21;
    compute_stats(xb_c, tb, wc, n, h, off5, cme0, cm20, cme1, cm21);
    __builtin_amdgcn_wave_barrier();
    v8f sme0, sm20, sme1, sm21;
    compute_stats(xb_s, tb, wc, n, h, off5, sme0, sm20, sme1, sm21);

    // Raw content center values (window rows 3..34, col 3+n).
    float xc[16];
    {
        const float* cp = xb_c + (3 + 8 * h) * 25 + 3 + n;
#pragma unroll
        for (int oc = 0; oc < 2; ++oc)
#pragma unroll
            for (int v = 0; v < 8; ++v)
                xc[oc * 8 + v] = cp[(16 * oc + v) * 25];
    }

    // out = (x - c_mean)/(sqrt(c_var)+eps) * (sqrt(s_var)+eps) + s_mean
    float* op = out + base + (long)((row0 + 8 * h) << 9) + col0 + n;
#pragma unroll
    for (int oc = 0; oc < 2; ++oc) {
#pragma unroll
        for (int v = 0; v < 8; ++v) {
            float cm = oc ? cme1[v] : cme0[v];
            float c2 = oc ? cm21[v] : cm20[v];
            float sm = oc ? sme1[v] : sme0[v];
            float s2 = oc ? sm21[v] : sm20[v];
            float cv = c2 - cm * cm;
            cv = cv > 0.0f ? cv : 0.0f;
            float sv = s2 - sm * sm;
            sv = sv > 0.0f ? sv : 0.0f;
            float cs = __builtin_amdgcn_sqrtf(cv) + 1e-5f;
            float ss = __builtin_amdgcn_sqrtf(sv) + 1e-5f;
            float o = (xc[oc * 8 + v] - cm) * (ss * __builtin_amdgcn_rcpf(cs)) + sm;
            op[(16 * oc + v) << 9] = o;
        }
    }
}

extern "C" void kernel_launch(void* const* d_in, const int* in_sizes, int n_in,
                              void* d_out, int out_size, void* d_ws, size_t ws_size,
                              hipStream_t stream) {
    (void)in_sizes; (void)n_in; (void)d_ws; (void)ws_size;
    const float* content = (const float*)d_in[0];
    const float* style   = (const float*)d_in[1];
    float* out = (float*)d_out;

    // out_size = 4*64*512*512 -> 131072 tiles of 32x16, 8 waves per block.
    const int total_tiles = out_size >> 9;
    const int blocks = (total_tiles + 7) / 8;
    adain_local_kernel<<<blocks, 256, 0, stream>>>(content, style, out, total_tiles);
}